// Process_3D_Motion_18760417149409
// MI455X (gfx1250) — compile-verified
//
#include <hip/hip_runtime.h>
#include <stdint.h>

// ---------------------------------------------------------------------------
// Forward kinematics, 55-joint SMPL-X tree, 22 driven body joints.
// Memory-bound (~47.5 MB @ 23.3 TB/s => ~2 us floor). Strategy:
//   phase 1: cooperative, fully-coalesced async copy of a 128-problem motion
//            tile (global -> LDS) via CDNA5 async-LDS engine (ASYNCcnt).
//   phase 2: one thread per problem walks the (constexpr, fully unrolled)
//            kinematic tree with register-resident 3x3 chain rotations,
//            writing the 165 output dwords back into its own LDS slot.
//   phase 3: cooperative, coalesced async store (LDS -> global).
// ---------------------------------------------------------------------------

#define NJ    55
#define NBODY 22
#define TILE  128              // problems per block (one problem per thread)
#define MDW   (NBODY * 9)      // 198 motion dwords per problem
#define ODW   (NJ * 3)         // 165 output dwords per problem

#define AS1 __attribute__((address_space(1)))
#define AS3 __attribute__((address_space(3)))

typedef int v4i __attribute__((ext_vector_type(4)));

// ---- CDNA5 async global<->LDS ops (ASYNCcnt), builtin with asm fallback ----

__device__ __forceinline__ void async_g2l_b128(void* lds_dst, const void* g_src) {
#if __has_builtin(__builtin_amdgcn_global_load_async_to_lds_b128)
  __builtin_amdgcn_global_load_async_to_lds_b128(
      (AS1 v4i*)g_src, (AS3 v4i*)lds_dst, /*offset=*/0, /*cpol=*/0);
#else
  // VDST holds the LDS byte offset; generic LDS address low 32 bits == offset.
  const unsigned lo = (unsigned)(unsigned long long)lds_dst;
  asm volatile("global_load_async_to_lds_b128 %0, %1, off"
               :: "v"(lo), "v"(g_src)
               : "memory");
#endif
}

__device__ __forceinline__ void async_l2g_b32(void* g_dst, const void* lds_src) {
#if __has_builtin(__builtin_amdgcn_global_store_async_from_lds_b32)
  __builtin_amdgcn_global_store_async_from_lds_b32(
      (AS1 int*)g_dst, (AS3 int*)lds_src, /*offset=*/0, /*cpol=*/0);
#else
  const unsigned lo = (unsigned)(unsigned long long)lds_src;
  asm volatile("global_store_async_from_lds_b32 %0, %1, off"
               :: "v"(g_dst), "v"(lo)
               : "memory");
#endif
}

__device__ __forceinline__ void wait_async0() {
#if __has_builtin(__builtin_amdgcn_s_wait_asynccnt)
  __builtin_amdgcn_s_wait_asynccnt(0);
#else
  asm volatile("s_wait_asynccnt 0" ::: "memory");
#endif
}

// ---------------------------------------------------------------------------

__global__ __launch_bounds__(TILE)
void Process_3D_Motion_fk_kernel(const float* __restrict__ motion,
                                 const float* __restrict__ Jrest,
                                 const int*   __restrict__ parents,
                                 float* __restrict__ out,
                                 int nprob) {
  // Per-problem slots, 198 dwords each. Slot stride 198 dwords is LDS
  // bank-conflict-free for per-thread access (gcd(198,64)=2 -> 32 distinct
  // banks across 32 lanes). Output (165 dwords) is written back into the
  // same slot: t[j] lands at dwords 3j..3j+2, while all motion reads at or
  // after step j start at dword 9j -> no overlap hazard within a thread.
  __shared__ __align__(16) float sm[TILE * MDW];
  __shared__ float rel[NJ * 3];

  const int tid = threadIdx.x;
  const long long base = (long long)blockIdx.x * TILE;
  if (base >= nprob) return;

  const float* gsrc = motion + base * MDW;

  // ---------------- phase 1: coalesced async stage, global -> LDS ----------
  constexpr int CHUNKS = TILE * MDW / 4;      // 6336 x 16B = 101376 B
  for (int c = tid; c < CHUNKS; c += TILE)
    async_g2l_b128(&sm[c * 4], gsrc + c * 4);

  // Rest-pose joint offsets (uniform across problems) -> LDS broadcast data.
  if (tid < NJ) {
    const int p = parents[tid];
#pragma unroll
    for (int k = 0; k < 3; ++k) {
      float v = Jrest[tid * 3 + k];
      if (p >= 0) v -= Jrest[p * 3 + k];
      rel[tid * 3 + k] = v;
    }
  }

  wait_async0();
  __syncthreads();

  // ---------------- phase 2: per-thread kinematic chain --------------------
  // Parent table and "effective chain-rotation index" (nearest ancestor with
  // a driven rotation; joints >= 22 have identity local rotation).
  constexpr int PAR[NJ] = {
      -1, 0, 0, 0, 1, 2, 3, 4, 5, 6, 7, 8, 9, 9, 9, 12, 13, 14, 16, 17, 18,
      19, 15, 15, 15, 20, 25, 26, 20, 28, 29, 20, 31, 32, 20, 34, 35, 20, 37,
      38, 21, 40, 41, 21, 43, 44, 21, 46, 47, 21, 49, 50, 21, 52, 53};
  constexpr int RIDX[NJ] = {
      0, 1, 2, 3, 4, 5, 6, 7, 8, 9, 10, 11, 12, 13, 14, 15, 16, 17, 18, 19,
      20, 21,                                    // body joints: themselves
      15, 15, 15,                                // 22..24 under joint 15
      20, 20, 20, 20, 20, 20, 20, 20, 20, 20, 20, 20, 20, 20, 20,  // 25..39
      21, 21, 21, 21, 21, 21, 21, 21, 21, 21, 21, 21, 21, 21, 21}; // 40..54

  float* slot = &sm[tid * MDW];

  float R[NBODY][9];   // chain rotations, register resident after full unroll
  float t[NJ][3];      // chain translations

#pragma unroll
  for (int k = 0; k < 9; ++k) R[0][k] = slot[k];
#pragma unroll
  for (int r = 0; r < 3; ++r) { t[0][r] = rel[r]; }
  // write t[0] after R[0] has been read (dwords 0..2 overlap motion dwords 0..8)
#pragma unroll
  for (int r = 0; r < 3; ++r) slot[r] = t[0][r];

#pragma unroll
  for (int j = 1; j < NJ; ++j) {
    const int p  = PAR[j];
    const int rp = RIDX[p];   // storage index of parent's chain rotation

    float rl[3];
#pragma unroll
    for (int k = 0; k < 3; ++k) rl[k] = rel[j * 3 + k];

    // t_j = R_chain[parent] * rel_j + t_parent
#pragma unroll
    for (int r = 0; r < 3; ++r) {
      const float v = fmaf(R[rp][r * 3 + 0], rl[0],
                      fmaf(R[rp][r * 3 + 1], rl[1],
                      fmaf(R[rp][r * 3 + 2], rl[2], t[p][r])));
      t[j][r] = v;
      slot[j * 3 + r] = v;        // safe: 3j+2 < 9j for j>=1
    }

    if (j < NBODY) {
      // R_j = R_parent * L_j   (local rotation from this thread's LDS slot)
      float L[9];
#pragma unroll
      for (int k = 0; k < 9; ++k) L[k] = slot[j * 9 + k];
#pragma unroll
      for (int r = 0; r < 3; ++r)
#pragma unroll
        for (int c = 0; c < 3; ++c)
          R[j][r * 3 + c] = fmaf(R[p][r * 3 + 0], L[0 * 3 + c],
                            fmaf(R[p][r * 3 + 1], L[1 * 3 + c],
                                 R[p][r * 3 + 2] * L[2 * 3 + c]));
    }
  }

  __syncthreads();

  // ---------------- phase 3: coalesced async store, LDS -> global ----------
  // Flat output dword f = p*ODW + d maps to LDS sm[p*MDW + d]. Track (p,d)
  // incrementally (TILE < ODW => at most one wrap per step, no div/mod).
  static_assert(TILE < ODW, "p/d tracking assumes TILE < ODW");
  float* gdst = out + base * ODW;
  int p = 0, d = tid;
  constexpr int TOT = TILE * ODW;   // 21120 dwords, uniform 165 iterations
  for (int f = tid; f < TOT; f += TILE) {
    async_l2g_b32(gdst + f, &sm[p * MDW + d]);
    d += TILE;
    if (d >= ODW) { d -= ODW; p += 1; }
  }
  // S_ENDPGM performs an implicit wait-idle, covering outstanding ASYNCcnt.
}

// ---------------------------------------------------------------------------

extern "C" void kernel_launch(void* const* d_in, const int* in_sizes, int n_in,
                              void* d_out, int out_size, void* d_ws, size_t ws_size,
                              hipStream_t stream) {
  const float* motion  = (const float*)d_in[0];   // (16,2048,22,3,3) f32
  const float* Jrest   = (const float*)d_in[1];   // (55,3) f32
  const int*   parents = (const int*)d_in[2];     // (55,) i32
  float*       out     = (float*)d_out;           // (16,2048,55,3) f32

  const int nprob  = in_sizes[0] / MDW;           // B*T = 32768
  const int blocks = (nprob + TILE - 1) / TILE;   // 256

  Process_3D_Motion_fk_kernel<<<blocks, TILE, 0, stream>>>(
      motion, Jrest, parents, out, nprob);
}